// LocalForceNet_37082747634270
// MI455X (gfx1250) — compile-verified
//
#include <hip/hip_runtime.h>

typedef __attribute__((ext_vector_type(2))) float v2f;
typedef __attribute__((ext_vector_type(8))) float v8f;

#define HID 128
#define CHID 32

// ---------------------------------------------------------------------------
// 1) Fuse weights:  Wf = W_node @ W_gcn[:128]   (128x128)
//                   Wc2 = W_coord @ W_gcn[128:160]  (2x128)
//                   bf  = b_node @ W_gcn[:128] + b_coord @ W_gcn[128:160]
// One workgroup of 128 threads; thread n owns output column n.
// ---------------------------------------------------------------------------
__global__ __launch_bounds__(128)
void fuse_weights_kernel(const float* __restrict__ Wn, const float* __restrict__ bn,
                         const float* __restrict__ Wc, const float* __restrict__ bc,
                         const float* __restrict__ Wg,
                         float* __restrict__ Wf, float* __restrict__ Wc2,
                         float* __restrict__ bf)
{
    const int n = threadIdx.x;
    float bacc = 0.f;
    for (int j = 0; j < HID; ++j)  bacc += bn[j] * Wg[j * HID + n];
    for (int j = 0; j < CHID; ++j) bacc += bc[j] * Wg[(HID + j) * HID + n];
    bf[n] = bacc;
    for (int c = 0; c < 2; ++c) {
        float acc = 0.f;
        for (int j = 0; j < CHID; ++j) acc += Wc[c * CHID + j] * Wg[(HID + j) * HID + n];
        Wc2[c * HID + n] = acc;
    }
    for (int i = 0; i < HID; ++i) {
        float acc = 0.f;
        for (int j = 0; j < HID; ++j) acc += Wn[i * HID + j] * Wg[j * HID + n];
        Wf[i * HID + n] = acc;
    }
}

// ---------------------------------------------------------------------------
// 2) Degree: deg[i] = 1 (self loop) + #edges with dst==i ; then dinv=rsqrt(deg)
// ---------------------------------------------------------------------------
__global__ void init_deg_kernel(float* __restrict__ deg, int N)
{
    int i = blockIdx.x * blockDim.x + threadIdx.x;
    if (i < N) deg[i] = 1.0f;
}

__global__ void deg_edges_kernel(const long long* __restrict__ dst,
                                 float* __restrict__ deg, int E)
{
    int e = blockIdx.x * blockDim.x + threadIdx.x;
    if (e < E) unsafeAtomicAdd(&deg[dst[e]], 1.0f);
}

__global__ void dinv_kernel(float* __restrict__ deg, int N)
{
    int i = blockIdx.x * blockDim.x + threadIdx.x;
    if (i < N) deg[i] = __frsqrt_rn(deg[i]);   // deg >= 1 always (self loop)
}

// ---------------------------------------------------------------------------
// 3) Zero the aggregation buffer (float4 stores)
// ---------------------------------------------------------------------------
__global__ void zero_kernel(float4* __restrict__ p, long long n4)
{
    long long i = (long long)blockIdx.x * blockDim.x + threadIdx.x;
    if (i < n4) p[i] = make_float4(0.f, 0.f, 0.f, 0.f);
}

// ---------------------------------------------------------------------------
// 4) Main GEMM (fp32 WMMA): xw[N,128] = X @ Wf + coords-rank2 + bf
//    Block = 128 threads = 4 waves; each wave owns one 16-row tile of output.
//    Wf staged in LDS in K-PAIRED layout: Wl[(k>>1)*256 + n*2 + (k&1)], so a
//    B fragment ((K,K+1) for one column) is one aligned ds_load_b64 straight
//    into an even VGPR pair — no packing moves, half the DS ops.
//    K stepped by 4 per v_wmma_f32_16x16x4_f32; 8 N-tiles => 8 accumulators.
//    Fragment layouts per ISA 7.12.2 (A: lane=M, vgpr0={K0|K2}; B: lane=N,
//    vgpr0={K0|K2}; C: vgpr v -> row v / v+8 per half-wave).
// ---------------------------------------------------------------------------
__global__ __launch_bounds__(128)
void gemm_xw_kernel(const float* __restrict__ X, const float* __restrict__ Wf,
                    const float* __restrict__ coords, const float* __restrict__ Wc2,
                    const float* __restrict__ bf, float* __restrict__ xw, int N)
{
    __shared__ float Wl[HID * HID];                 // 64 KB, K-paired layout
    const int tid  = threadIdx.x;
    const int lane = tid & 31;
    const int wv   = tid >> 5;                      // wave in block: 0..3

    // cooperative load of Wf into LDS with pair-swizzle (global side coalesced)
    for (int i = tid; i < HID * HID; i += 128) {
        const int k = i >> 7;
        const int n = i & 127;
        Wl[(k >> 1) * 256 + n * 2 + (k & 1)] = Wf[i];
    }
    __syncthreads();

    const int mtile = blockIdx.x * 4 + wv;
    const int m0 = mtile * 16;
    if (m0 >= N) return;                            // whole-wave uniform exit

    const int mrow  = lane & 15;                    // M (for A) / N (for B,C)
    const int khalf = (lane >> 4) << 1;             // 0 for lanes 0-15, 2 for 16-31
    const int kpair = khalf >> 1;                   // 0 or 1 (pair index offset)
    const int mbase = (lane >> 4) << 3;             // C rows: v or v+8

    v8f acc[8] = {};

    const float* Xrow = X + (size_t)(m0 + mrow) * HID;
    for (int k0 = 0; k0 < HID; k0 += 4) {
        v2f a = *(const v2f*)(Xrow + k0 + khalf);    // A frag: (M=mrow, K=kk,kk+1)
        const float* B0 = &Wl[((k0 >> 1) + kpair) * 256 + mrow * 2];
#pragma unroll
        for (int nt = 0; nt < 8; ++nt) {
            v2f b = *(const v2f*)(B0 + nt * 32);     // one ds_load_b64
            acc[nt] = __builtin_amdgcn_wmma_f32_16x16x4_f32(
                false, a, false, b, (short)0, acc[nt], false, false);
        }
    }

    // epilogue: rank-2 coordinate term + fused bias, then store
    float cx[8], cy[8];
#pragma unroll
    for (int v = 0; v < 8; ++v) {
        const float2 c = ((const float2*)coords)[m0 + mbase + v];
        cx[v] = c.x; cy[v] = c.y;
    }
#pragma unroll
    for (int nt = 0; nt < 8; ++nt) {
        const int n = nt * 16 + mrow;
        const float w0 = Wc2[n], w1 = Wc2[HID + n], bb = bf[n];
#pragma unroll
        for (int v = 0; v < 8; ++v) {
            const float val = acc[nt][v] + cx[v] * w0 + cy[v] * w1 + bb;
            xw[(size_t)(m0 + mbase + v) * HID + n] = val;
        }
    }
}

// ---------------------------------------------------------------------------
// 5) Edge scatter: one wave32 per edge. Gather 512B row of xw[src] (L2-resident
//    since xw is 51MB < 192MB L2), scale by dinv[s]*dinv[d], hardware fp32
//    atomics into agg[dst]. Self-loops handled in the head kernel (no atomics).
// ---------------------------------------------------------------------------
__global__ __launch_bounds__(256)
void scatter_kernel(const long long* __restrict__ src, const long long* __restrict__ dst,
                    const float* __restrict__ xw, const float* __restrict__ dinv,
                    float* __restrict__ agg, int E)
{
    const int w    = (blockIdx.x * blockDim.x + threadIdx.x) >> 5;
    const int lane = threadIdx.x & 31;
    if (w >= E) return;
    const long long s = src[w];
    const long long d = dst[w];
    const float nrm = dinv[s] * dinv[d];
    const float4 v = ((const float4*)(xw + (size_t)s * HID))[lane];
    float* a = agg + (size_t)d * HID + lane * 4;
    unsafeAtomicAdd(a + 0, v.x * nrm);
    unsafeAtomicAdd(a + 1, v.y * nrm);
    unsafeAtomicAdd(a + 2, v.z * nrm);
    unsafeAtomicAdd(a + 3, v.w * nrm);
}

// ---------------------------------------------------------------------------
// 6) Head: one wave per node.  h2 = relu(agg + xw*dinv^2 + b_gcn)
//          out = h2 @ W_head + b_head   (128 -> 2, shuffle reduction)
// ---------------------------------------------------------------------------
__global__ __launch_bounds__(256)
void head_kernel(const float* __restrict__ agg, const float* __restrict__ xw,
                 const float* __restrict__ dinv, const float* __restrict__ bg,
                 const float* __restrict__ Wh, const float* __restrict__ bh,
                 float* __restrict__ out, int N)
{
    const int node = (blockIdx.x * blockDim.x + threadIdx.x) >> 5;
    const int lane = threadIdx.x & 31;
    if (node >= N) return;
    const float di = dinv[node];
    const float sn = di * di;                      // self-loop norm
    const int f0 = lane * 4;
    const float4 a4 = *(const float4*)(agg + (size_t)node * HID + f0);
    const float4 x4 = *(const float4*)(xw  + (size_t)node * HID + f0);
    const float4 b4 = *(const float4*)(bg + f0);
    float h[4];
    h[0] = fmaxf(a4.x + x4.x * sn + b4.x, 0.f);
    h[1] = fmaxf(a4.y + x4.y * sn + b4.y, 0.f);
    h[2] = fmaxf(a4.z + x4.z * sn + b4.z, 0.f);
    h[3] = fmaxf(a4.w + x4.w * sn + b4.w, 0.f);
    float p0 = 0.f, p1 = 0.f;
#pragma unroll
    for (int j = 0; j < 4; ++j) {
        p0 += h[j] * Wh[(f0 + j) * 2 + 0];
        p1 += h[j] * Wh[(f0 + j) * 2 + 1];
    }
#pragma unroll
    for (int off = 16; off >= 1; off >>= 1) {      // wave32 reduction
        p0 += __shfl_xor(p0, off, 32);
        p1 += __shfl_xor(p1, off, 32);
    }
    if (lane == 0) {
        out[(size_t)node * 2 + 0] = p0 + bh[0];
        out[(size_t)node * 2 + 1] = p1 + bh[1];
    }
}

// ---------------------------------------------------------------------------
extern "C" void kernel_launch(void* const* d_in, const int* in_sizes, int n_in,
                              void* d_out, int out_size, void* d_ws, size_t ws_size,
                              hipStream_t stream)
{
    const float*     x      = (const float*)d_in[0];
    const long long* eidx   = (const long long*)d_in[1];   // int64 [2,E]
    const float*     coords = (const float*)d_in[2];
    const float*     Wn     = (const float*)d_in[3];
    const float*     bn     = (const float*)d_in[4];
    const float*     Wc     = (const float*)d_in[5];
    const float*     bc     = (const float*)d_in[6];
    const float*     Wg     = (const float*)d_in[7];
    const float*     bg     = (const float*)d_in[8];
    const float*     Wh     = (const float*)d_in[9];
    const float*     bh     = (const float*)d_in[10];
    float*           out    = (float*)d_out;

    const int N = in_sizes[0] / HID;          // 100000
    const int E = in_sizes[1] / 2;            // 1600000
    const long long* src = eidx;
    const long long* dst = eidx + E;

    // workspace carve-up
    char* ws = (char*)d_ws;
    float* Wf   = (float*)ws; ws += (size_t)HID * HID * 4;          // 64 KB
    float* Wc2  = (float*)ws; ws += 2 * HID * 4;                    // 1 KB
    float* bf   = (float*)ws; ws += HID * 4;                        // 0.5 KB
    float* dinv = (float*)ws; ws += (((size_t)N * 4) + 255) & ~(size_t)255;
    float* xw   = (float*)ws; ws += (size_t)N * HID * 4;            // 51.2 MB
    float* agg  = (float*)ws;                                       // 51.2 MB

    // 1) fuse weights (tiny)
    fuse_weights_kernel<<<1, 128, 0, stream>>>(Wn, bn, Wc, bc, Wg, Wf, Wc2, bf);

    // 2) degrees -> dinv  (reuse buffer in place)
    init_deg_kernel<<<(N + 255) / 256, 256, 0, stream>>>(dinv, N);
    deg_edges_kernel<<<(E + 255) / 256, 256, 0, stream>>>(dst, dinv, E);
    dinv_kernel<<<(N + 255) / 256, 256, 0, stream>>>(dinv, N);

    // 3) zero agg
    {
        const long long n4 = (long long)N * HID / 4;
        zero_kernel<<<(unsigned)((n4 + 255) / 256), 256, 0, stream>>>((float4*)agg, n4);
    }

    // 4) fused projection GEMM (WMMA fp32)
    {
        const int mtiles = (N + 15) / 16;
        gemm_xw_kernel<<<(mtiles + 3) / 4, 128, 0, stream>>>(x, Wf, coords, Wc2, bf, xw, N);
    }

    // 5) edge scatter (wave per edge)
    scatter_kernel<<<(E * 32 + 255) / 256, 256, 0, stream>>>(src, dst, xw, dinv, agg, E);

    // 6) relu + head
    head_kernel<<<((size_t)N * 32 + 255) / 256, 256, 0, stream>>>(agg, xw, dinv, bg, Wh, bh, out, N);
}